// ScaledDotProductAttention_50981261803629
// MI455X (gfx1250) — compile-verified
//
#include <hip/hip_runtime.h>

// ============================================================================
// ScaledDotProductAttention (B=131072, T=8, H=32), f32, MI455X / gfx1250.
//
// Restructured math (exact up to f32 reassociation):
//   u[b]       = h_dec[b] @ Wc + bu,  Wc = Wq^T @ Wk,  bu = bq @ Wk
//   s[b,t]     = (h_enc[b,t] . u[b]) / sqrt(32)        (q.bk cancels in softmax)
//   att        = softmax_t(s)
//   a[b]       = sum_t att[b,t] * h_enc[b,t]
//   context[b] = a[b] @ Wv^T + bv                      (since sum_t att = 1)
//
// GEMMs use V_WMMA_F32_16X16X4_F32 (exact f32). One wave owns 16 batch rows.
// Cross-lane reductions use single-instruction DS_SWIZZLE_B32 XOR butterflies.
// ============================================================================

#define HIDDEN 32
#define HT 8
#define ROWS_PER_WAVE 16
#define WAVES_PER_BLOCK 4
#define ROWS_PER_BLOCK (ROWS_PER_WAVE * WAVES_PER_BLOCK)
#define LDS_STRIDE 33   // pad to avoid LDS bank conflicts on column reads

typedef __attribute__((ext_vector_type(2))) float v2f;
typedef __attribute__((ext_vector_type(8))) float v8f;

__device__ __forceinline__ v8f wmma_f32_16x16x4(v2f a, v2f b, v8f c) {
  // 8-arg pattern: (neg_a, A, neg_b, B, c_mod, C, reuse_a, reuse_b)
  return __builtin_amdgcn_wmma_f32_16x16x4_f32(false, a, false, b, (short)0, c,
                                               false, false);
}

// ds_swizzle_b32 group-of-32 XOR exchange: offset = (xor<<10)|(or<<5)|and.
template <int IMM>
__device__ __forceinline__ float swz_add(float v) {
  return v + __int_as_float(__builtin_amdgcn_ds_swizzle(__float_as_int(v), IMM));
}

// Full 32-lane sum, result replicated in every lane (5 ds_swizzle + 5 adds).
__device__ __forceinline__ float wave_sum32(float v) {
  v = swz_add<0x041F>(v);  // xor 1
  v = swz_add<0x081F>(v);  // xor 2
  v = swz_add<0x101F>(v);  // xor 4
  v = swz_add<0x201F>(v);  // xor 8
  v = swz_add<0x401F>(v);  // xor 16 (SWAPX16)
  return v;
}

// ---------------------------------------------------------------------------
// Prep kernel: ws[0..1023]  = WcT  (layout [h][hin] so B[k=hin][n=h] pairs
//                                   (k,k+1) are contiguous for v2f loads)
//              ws[1024..55] = bu[h] = sum_d bq[d]*Wk[d][h]
// ---------------------------------------------------------------------------
__global__ void attn_prep(const float* __restrict__ Wq,
                          const float* __restrict__ bq,
                          const float* __restrict__ Wk,
                          float* __restrict__ ws) {
  const int tid = threadIdx.x;                 // 256 threads, 1 block
  for (int i = tid; i < HIDDEN * HIDDEN; i += 256) {
    const int h = i >> 5, hin = i & 31;
    float acc = 0.f;
#pragma unroll
    for (int d = 0; d < HIDDEN; ++d)
      acc += Wq[d * HIDDEN + hin] * Wk[d * HIDDEN + h];
    ws[i] = acc;                               // WcT[h][hin]
  }
  if (tid < HIDDEN) {
    float acc = 0.f;
#pragma unroll
    for (int d = 0; d < HIDDEN; ++d) acc += bq[d] * Wk[d * HIDDEN + tid];
    ws[HIDDEN * HIDDEN + tid] = acc;           // bu[h]
  }
}

// ---------------------------------------------------------------------------
// Main kernel: 128 threads = 4 waves; each wave owns 16 batch rows.
// ---------------------------------------------------------------------------
__global__ __launch_bounds__(128) void attn_main(
    const float* __restrict__ h_enc, const float* __restrict__ h_dec,
    const float* __restrict__ Wv, const float* __restrict__ bv,
    const float* __restrict__ WcT, const float* __restrict__ bu,
    float* __restrict__ out, int B) {
  __shared__ float lds_u[WAVES_PER_BLOCK][ROWS_PER_WAVE * LDS_STRIDE];
  __shared__ float lds_a[WAVES_PER_BLOCK][ROWS_PER_WAVE * LDS_STRIDE];

  const int tid  = threadIdx.x;
  const int wave = tid >> 5;
  const int lane = tid & 31;
  const int n    = lane & 15;   // N-column within tile / M-row for A operand
  const int kh   = lane >> 4;   // K-half select (A/B f32 layout)

  const int row0 = blockIdx.x * ROWS_PER_BLOCK + wave * ROWS_PER_WAVE;
  if (row0 + ROWS_PER_WAVE > B) return;  // B = 131072 is a multiple of 64

  float* __restrict__ lu = lds_u[wave];
  float* __restrict__ la = lds_a[wave];

  // ---- Preload both B operands into VGPRs for the whole kernel.
  // f32 B 4x16 layout: lane<16 -> n=lane, VGPR0=k0,VGPR1=k1;
  //                    lane>=16 -> n=lane-16, VGPR0=k2,VGPR1=k3.
  v2f Bu[2][8], Bv[2][8];
#pragma unroll
  for (int nt = 0; nt < 2; ++nt) {
#pragma unroll
    for (int kk = 0; kk < 8; ++kk) {
      const int col = 16 * nt + n;
      const int k0  = 4 * kk + 2 * kh;
      Bu[nt][kk] = *(const v2f*)(WcT + col * HIDDEN + k0);  // B[k][n]=Wc[k][n]
      Bv[nt][kk] = *(const v2f*)(Wv  + col * HIDDEN + k0);  // B[k][n]=Wv[n][k]
    }
  }

  // ---- GEMM 1: u[16][32] = h_dec_tile[16][32] @ Wc + bu  (WMMA f32 16x16x4)
  v8f cu0, cu1;
  {
    const float b0 = bu[n], b1 = bu[16 + n];
    cu0 = (v8f){b0, b0, b0, b0, b0, b0, b0, b0};
    cu1 = (v8f){b1, b1, b1, b1, b1, b1, b1, b1};
  }
#pragma unroll
  for (int kk = 0; kk < 8; ++kk) {
    // A 16x4 f32 layout: lane<16 m=lane holds K=4kk+{0,1}; lane>=16 K=4kk+{2,3}
    v2f a = *(const v2f*)(h_dec + (size_t)(row0 + n) * HIDDEN + 4 * kk + 2 * kh);
    cu0 = wmma_f32_16x16x4(a, Bu[0][kk], cu0);
    cu1 = wmma_f32_16x16x4(a, Bu[1][kk], cu1);
  }
  // Scatter C-layout (VGPR j -> row j / j+8, lane -> col) into LDS [row][h].
#pragma unroll
  for (int j = 0; j < 8; ++j) {
    const int m = j + 8 * kh;
    lu[m * LDS_STRIDE + n]      = cu0[j];
    lu[m * LDS_STRIDE + 16 + n] = cu1[j];
  }

  // ---- Stage 2: per-row softmax attention; lanes = hidden dim (coalesced).
  const float* __restrict__ encw = h_enc + (size_t)row0 * (HT * HIDDEN);
  const float kScale = 0.17677669529663687f * 1.4426950408889634f; // 1/sqrt(32)*log2(e)
#pragma unroll 1
  for (int r = 0; r < ROWS_PER_WAVE; ++r) {
    const float* __restrict__ ep = encw + (size_t)r * (HT * HIDDEN);
    if (r + 1 < ROWS_PER_WAVE)   // pull next 1KB row into cache (global_prefetch_b8)
      __builtin_prefetch(ep + HT * HIDDEN + lane * 8, 0, 3);

    float e[HT];
#pragma unroll
    for (int t = 0; t < HT; ++t) e[t] = ep[t * HIDDEN + lane];

    const float uh = lu[r * LDS_STRIDE + lane];
    float s[HT];
    // 8 timestep dots: per-lane product + single-instruction swizzle butterflies.
#pragma unroll
    for (int t = 0; t < HT; ++t) s[t] = wave_sum32(e[t] * uh);
    // Softmax over t (replicated in every lane).
    float mx = s[0];
#pragma unroll
    for (int t = 1; t < HT; ++t) mx = fmaxf(mx, s[t]);
    float p[HT], den = 0.f;
#pragma unroll
    for (int t = 0; t < HT; ++t) {
      p[t] = __builtin_amdgcn_exp2f((s[t] - mx) * kScale);  // native v_exp_f32
      den += p[t];
    }
    const float inv = 1.0f / den;
    float ah = 0.f;
#pragma unroll
    for (int t = 0; t < HT; ++t) ah += p[t] * e[t];
    la[r * LDS_STRIDE + lane] = ah * inv;      // a[r][h]
  }

  // ---- GEMM 2: context[16][32] = a[16][32] @ Wv^T + bv
  v8f cc0, cc1;
  {
    const float b0 = bv[n], b1 = bv[16 + n];
    cc0 = (v8f){b0, b0, b0, b0, b0, b0, b0, b0};
    cc1 = (v8f){b1, b1, b1, b1, b1, b1, b1, b1};
  }
#pragma unroll
  for (int kk = 0; kk < 8; ++kk) {
    const int k0 = 4 * kk + 2 * kh;
    v2f a;
    a.x = la[n * LDS_STRIDE + k0];
    a.y = la[n * LDS_STRIDE + k0 + 1];
    cc0 = wmma_f32_16x16x4(a, Bv[0][kk], cc0);
    cc1 = wmma_f32_16x16x4(a, Bv[1][kk], cc1);
  }
#pragma unroll
  for (int j = 0; j < 8; ++j) {
    const int m = j + 8 * kh;
    out[(size_t)(row0 + m) * HIDDEN + n]      = cc0[j];
    out[(size_t)(row0 + m) * HIDDEN + 16 + n] = cc1[j];
  }
}

// ---------------------------------------------------------------------------
extern "C" void kernel_launch(void* const* d_in, const int* in_sizes, int n_in,
                              void* d_out, int out_size, void* d_ws,
                              size_t ws_size, hipStream_t stream) {
  const float* h_enc = (const float*)d_in[0];
  const float* h_dec = (const float*)d_in[1];
  const float* Wq    = (const float*)d_in[2];
  const float* bq    = (const float*)d_in[3];
  const float* Wk    = (const float*)d_in[4];
  const float* bk    = (const float*)d_in[5];  // cancels inside softmax
  const float* Wv    = (const float*)d_in[6];
  const float* bv    = (const float*)d_in[7];
  (void)bk; (void)n_in; (void)out_size; (void)ws_size;

  float* out = (float*)d_out;
  float* ws  = (float*)d_ws;                   // 1024 floats WcT + 32 floats bu

  const int B = in_sizes[1] / HIDDEN;          // h_dec is [B, 32]

  attn_prep<<<1, 256, 0, stream>>>(Wq, bq, Wk, ws);

  const int blocks = (B + ROWS_PER_BLOCK - 1) / ROWS_PER_BLOCK;
  attn_main<<<blocks, 128, 0, stream>>>(h_enc, h_dec, Wv, bv, ws,
                                        ws + HIDDEN * HIDDEN, out, B);
}